// SparseConv2d_38405597561045
// MI455X (gfx1250) — compile-verified
//
#include <hip/hip_runtime.h>
#include <stdint.h>

typedef __attribute__((ext_vector_type(2))) float v2f;
typedef __attribute__((ext_vector_type(8))) float v8f;

#define CIN   256
#define COUT  256
#define HH    56
#define WW    56
#define HW    3136              // 56*56
#define NB    32
#define KTOT  2304              // CIN*9
#define NPIX  (NB * HW)         // 100352

#define BM    64                // couts per block
#define BN    256               // pixels per block
#define BK    16                // channels per stage (8 k-pair rows)
#define LDA2  80                // A float2 row stride (64 + 16): 32-bank shift/row
#define LDB2  272               // B float2 row stride (256 + 16): 32-bank shift/row
#define NSTAGE (9 * (CIN / BK)) // 9 taps * 16 channel-slabs = 144

#if defined(__gfx1250__) && __has_builtin(__builtin_amdgcn_global_load_async_to_lds_b32)
#define HAS_ASYNC 1
#else
#define HAS_ASYNC 0
#endif

#if HAS_ASYNC
typedef __attribute__((address_space(1))) int* gas_i32_t;  // global
typedef __attribute__((address_space(3))) int* las_i32_t;  // LDS
// async copy: one 32-bit element, global -> this lane's LDS slot (no VGPR data)
static __device__ __forceinline__ void cp_async_b32(const void* gsrc, void* ldst) {
    __builtin_amdgcn_global_load_async_to_lds_b32(
        (gas_i32_t)(uintptr_t)gsrc,
        (las_i32_t)(uint32_t)(uintptr_t)ldst,
        0, 0);
}
#endif

static __device__ __forceinline__ void async_wait_all() {
#if HAS_ASYNC
#if __has_builtin(__builtin_amdgcn_s_wait_asynccnt)
    __builtin_amdgcn_s_wait_asynccnt(0);
#else
    asm volatile("s_wait_asynccnt 0x0" ::: "memory");
#endif
#endif
}

__global__ __launch_bounds__(256) void conv3x3_wmma_f32(
    const float* __restrict__ Xg,     // [NB, CIN, 56, 56]
    const float* __restrict__ Wg,     // [COUT, CIN, 3, 3]
    const float* __restrict__ Bias,   // [COUT]
    float* __restrict__ Out)          // [NB, COUT, 56, 56]
{
    // K-pair-major LDS: element (k, col) = s?2[(k>>1)*LD?2 + col][k&1]
    // -> every WMMA fragment is ONE naturally aligned ds_load_b64.
    __shared__ v2f sA2[2][(BK / 2) * LDA2];
    __shared__ v2f sB2[2][(BK / 2) * LDB2];

    const int tid  = threadIdx.x;
    const int lane = tid & 31;
    const int wave = tid >> 5;
    const int m0   = (wave & 1) * 32;    // 2 waves along M, 32 rows each
    const int n0   = (wave >> 1) * 64;   // 4 waves along N, 64 cols each

    const int mbase = blockIdx.y * BM;
    const int nbase = blockIdx.x * BN;

    // ---- per-thread staging geometry (computed once; no divisions in loop) ----
    const int ml    = tid & 63;
    const int kp0   = tid >> 6;                       // 0..3
    const int wrowA = (mbase + ml) * KTOT;
    const int nl  = tid;                              // one pixel per thread
    const int n   = nbase + nl;
    const int bb  = n / HW;
    const int p   = n - bb * HW;
    const int yy  = p / WW;
    const int xx  = p - yy * WW;
    const int pixbase = bb * CIN * HW + (yy - 1) * WW + (xx - 1);

    v8f c00 = {}, c01 = {}, c02 = {}, c03 = {};
    v8f c10 = {}, c11 = {}, c12 = {}, c13 = {};

    const int lmod = lane & 15;
    const int kb   = (lane >> 4) << 1;   // lanes 0-15: K 0,1 ; lanes 16-31: K 2,3

    auto load_stage = [&](int st, int dst) {
        const int rs = st >> 4;              // tap index 0..8 (uniform)
        const int ct = (st & 15) << 4;       // channel slab base (uniform)
        const int r  = rs / 3;
        const int s  = rs - r * 3;
        // ---- A: 16x64 weight slab (always in bounds) ----
        #pragma unroll
        for (int i = 0; i < 2; ++i) {
            const int kp   = kp0 + 4 * i;                 // k-pair row 0..7
            const int aoff = wrowA + (ct + 2 * kp) * 9 + rs;
            v2f* slot = &sA2[dst][kp * LDA2 + ml];
#if HAS_ASYNC
            cp_async_b32(&Wg[aoff],     (float*)slot);
            cp_async_b32(&Wg[aoff + 9], (float*)slot + 1);
#else
            v2f w2;
            w2.x = Wg[aoff];
            w2.y = Wg[aoff + 9];
            *slot = w2;
#endif
        }
        // ---- B: 16x256 slab of X shifted by (r-1, s-1); zero outside image ----
        {
            const bool ok = ((unsigned)(yy + r - 1) < (unsigned)HH) &&
                            ((unsigned)(xx + s - 1) < (unsigned)WW);
            const int goff = pixbase + r * WW + s + ct * HW;
#if HAS_ASYNC
            if (ok) {
                #pragma unroll
                for (int t = 0; t < 8; ++t) {
                    v2f* slot = &sB2[dst][t * LDB2 + nl];
                    cp_async_b32(&Xg[goff + (2 * t) * HW],     (float*)slot);
                    cp_async_b32(&Xg[goff + (2 * t + 1) * HW], (float*)slot + 1);
                }
            } else {
                const v2f z = {0.0f, 0.0f};
                #pragma unroll
                for (int t = 0; t < 8; ++t)
                    sB2[dst][t * LDB2 + nl] = z;
            }
#else
            #pragma unroll
            for (int t = 0; t < 8; ++t) {
                v2f x2 = {0.0f, 0.0f};
                if (ok) {
                    x2.x = Xg[goff + (2 * t) * HW];
                    x2.y = Xg[goff + (2 * t + 1) * HW];
                }
                sB2[dst][t * LDB2 + nl] = x2;
            }
#endif
        }
    };

    load_stage(0, 0);

    int buf = 0;
    for (int st = 0; st < NSTAGE; ++st) {
        async_wait_all();                     // stage-`st` async fills drained
        __syncthreads();                      // buffer `buf` ready; other free
        if (st + 1 < NSTAGE)
            load_stage(st + 1, buf ^ 1);      // fire-and-forget under WMMA

        #pragma unroll
        for (int kk = 0; kk < BK; kk += 4) {
            const int rowA = ((kk + kb) >> 1) * LDA2;
            const int rowB = ((kk + kb) >> 1) * LDB2;
            const v2f aLo = sA2[buf][rowA + m0 + lmod];        // rows m0..m0+15
            const v2f aHi = sA2[buf][rowA + m0 + 16 + lmod];   // rows m0+16..+31
            const v2f b0  = sB2[buf][rowB + n0 + lmod];
            const v2f b1  = sB2[buf][rowB + n0 + 16 + lmod];
            const v2f b2  = sB2[buf][rowB + n0 + 32 + lmod];
            const v2f b3  = sB2[buf][rowB + n0 + 48 + lmod];
            c00 = __builtin_amdgcn_wmma_f32_16x16x4_f32(false, aLo, false, b0, (short)0, c00, false, false);
            c01 = __builtin_amdgcn_wmma_f32_16x16x4_f32(false, aLo, false, b1, (short)0, c01, false, false);
            c02 = __builtin_amdgcn_wmma_f32_16x16x4_f32(false, aLo, false, b2, (short)0, c02, false, false);
            c03 = __builtin_amdgcn_wmma_f32_16x16x4_f32(false, aLo, false, b3, (short)0, c03, false, false);
            c10 = __builtin_amdgcn_wmma_f32_16x16x4_f32(false, aHi, false, b0, (short)0, c10, false, false);
            c11 = __builtin_amdgcn_wmma_f32_16x16x4_f32(false, aHi, false, b1, (short)0, c11, false, false);
            c12 = __builtin_amdgcn_wmma_f32_16x16x4_f32(false, aHi, false, b2, (short)0, c12, false, false);
            c13 = __builtin_amdgcn_wmma_f32_16x16x4_f32(false, aHi, false, b3, (short)0, c13, false, false);
        }
        buf ^= 1;
    }

    // ---- epilogue: C/D layout VGPR r -> (M=r, lanes 0-15) / (M=r+8, lanes 16-31) ----
    const int mrow_off = (lane >> 4) << 3;            // 0 or 8
    auto store_tile = [&](const v8f& acc, int mh, int j) {
        const int cbase = mbase + m0 + mh * 16 + mrow_off;
        const int nn    = nbase + n0 + j * 16 + lmod;
        const int ob    = nn / HW;
        const int op    = nn - ob * HW;
        const int base  = (ob * COUT + cbase) * HW + op;
        #pragma unroll
        for (int r = 0; r < 8; ++r)
            Out[base + r * HW] = acc[r] + Bias[cbase + r];
    };
    store_tile(c00, 0, 0); store_tile(c01, 0, 1);
    store_tile(c02, 0, 2); store_tile(c03, 0, 3);
    store_tile(c10, 1, 0); store_tile(c11, 1, 1);
    store_tile(c12, 1, 2); store_tile(c13, 1, 3);
}

extern "C" void kernel_launch(void* const* d_in, const int* in_sizes, int n_in,
                              void* d_out, int out_size, void* d_ws, size_t ws_size,
                              hipStream_t stream) {
    const float* x    = (const float*)d_in[0];   // [32,256,56,56]
    const float* w    = (const float*)d_in[1];   // [256,256,3,3]
    const float* bias = (const float*)d_in[2];   // [256]
    float*       out  = (float*)d_out;           // [32,256,56,56]

    dim3 grid(NPIX / BN, COUT / BM);             // (392, 4)
    dim3 block(256);                             // 8 wave32s
    conv3x3_wmma_f32<<<grid, block, 0, stream>>>(x, w, bias, out);
}